// GrwSmoothingLoss_83227876261914
// MI455X (gfx1250) — compile-verified
//
#include <hip/hip_runtime.h>

#define B_SZ 256
#define T_SZ 8
#define K_SZ 128
#define NPERM 1000
#define NPERM_PAD 1008   // padded to multiple of 16 for WMMA tiles
#define ALPHA 0.5f

typedef __attribute__((ext_vector_type(2))) float v2f;
typedef __attribute__((ext_vector_type(8))) float v8f;

// ---------------------------------------------------------------------------
// Kernel 1: per-batch Gram matrix G_b = Z_b * Z_b^T  (8x8), Z_b is 8x128.
// One block (64 threads) per batch; thread e computes G[e>>3][e&7].
// ---------------------------------------------------------------------------
__global__ void gram_kernel(const float* __restrict__ Z, float* __restrict__ G) {
    int b = blockIdx.x;
    int e = threadIdx.x;                  // 0..63
    int i = e >> 3, j = e & 7;
    const float4* zi = (const float4*)(Z + (size_t)(b * T_SZ + i) * K_SZ);
    const float4* zj = (const float4*)(Z + (size_t)(b * T_SZ + j) * K_SZ);
    float acc = 0.f;
#pragma unroll 8
    for (int k = 0; k < K_SZ / 4; ++k) {
        float4 x = zi[k], y = zj[k];
        acc += x.x * y.x + x.y * y.y + x.z * y.z + x.w * y.w;
    }
    G[(size_t)b * 64 + e] = acc;
}

// ---------------------------------------------------------------------------
// Kernel 2: build Q_p (8x8) per permutation: Q_p[a][c] = M[inv(a)][inv(c)],
// M = C^T C for the 6x8 second-difference operator C. Pad perms -> 0.
// ---------------------------------------------------------------------------
__device__ __forceinline__ float d2c(int t, int i) {
    float c = 0.f;
    if (i == t)     c += 1.f;
    if (i == t + 1) c -= 2.f;
    if (i == t + 2) c += 1.f;
    return c;
}

__global__ void build_q_kernel(const int* __restrict__ perm, float* __restrict__ Q) {
    __shared__ int inv[T_SZ];
    int p = blockIdx.x;                   // 0..1007
    int e = threadIdx.x;                  // 0..63
    if (p < NPERM && e < T_SZ) inv[perm[p * T_SZ + e]] = e;
    __syncthreads();
    float val = 0.f;
    if (p < NPERM) {
        int ia = inv[e >> 3];
        int jc = inv[e & 7];
#pragma unroll
        for (int t = 0; t < T_SZ - 2; ++t) val += d2c(t, ia) * d2c(t, jc);
    }
    Q[(size_t)p * 64 + e] = val;
}

// ---------------------------------------------------------------------------
// Kernel 3: logits[b][p] = -0.5 * dot(Gflat[b], Qflat[p]) as a WMMA GEMM.
// A = G (256x64), B = Q^T (64x1008). One wave per 16x16 tile, K=64 as
// 16 steps of V_WMMA_F32_16X16X4_F32 (exact fp32).
// ---------------------------------------------------------------------------
__global__ void logits_wmma_kernel(const float* __restrict__ G,
                                   const float* __restrict__ Q,
                                   float* __restrict__ logits) {
    int tn = blockIdx.x;                  // 0..62  (perm tiles)
    int tm = blockIdx.y;                  // 0..15  (batch tiles)
    int l    = threadIdx.x;               // 0..31 (one wave32)
    int half = l >> 4;                    // 0: K=+0/+1  1: K=+2/+3
    int lm   = l & 15;

    const float* Arow = G + (size_t)(tm * 16 + lm) * 64;   // A row = batch
    const float* Brow = Q + (size_t)(tn * 16 + lm) * 64;   // B col = perm (Q row)

    v8f c = {};
#pragma unroll
    for (int kk = 0; kk < 16; ++kk) {
        int kb = kk * 4 + half * 2;
        v2f a, bf;
        a.x  = Arow[kb];     a.y  = Arow[kb + 1];
        bf.x = Brow[kb];     bf.y = Brow[kb + 1];
        c = __builtin_amdgcn_wmma_f32_16x16x4_f32(
                /*neg_a=*/false, a, /*neg_b=*/false, bf,
                /*c_mod=*/(short)0, c, /*reuse_a=*/false, /*reuse_b=*/false);
    }

    // D layout: VGPR v -> M = v (+8 for lanes 16-31), N = lane&15
    float* out = logits + (size_t)(tm * 16 + half * 8) * NPERM_PAD + tn * 16 + lm;
#pragma unroll
    for (int v = 0; v < 8; ++v) out[(size_t)v * NPERM_PAD] = -0.5f * c[v];
}

// ---------------------------------------------------------------------------
// Kernel 4: per-batch logsumexp(logits) - logits[0] + ALPHA * V_b, where
// V_b = 0.5 * sum_t (G[t][t] - 2 G[t][t+1] + G[t+1][t+1])  (first-diff energy).
// ---------------------------------------------------------------------------
__global__ void rowreduce_kernel(const float* __restrict__ logits,
                                 const float* __restrict__ G,
                                 float* __restrict__ perB) {
    __shared__ float sdata[128];
    int b = blockIdx.x;
    int t = threadIdx.x;                  // 0..127
    const float* row = logits + (size_t)b * NPERM_PAD;

    float m = -3.402823466e38f;
    for (int p = t; p < NPERM; p += 128) m = fmaxf(m, row[p]);
    sdata[t] = m; __syncthreads();
    for (int s = 64; s > 0; s >>= 1) {
        if (t < s) sdata[t] = fmaxf(sdata[t], sdata[t + s]);
        __syncthreads();
    }
    float rowmax = sdata[0]; __syncthreads();

    float sum = 0.f;
    for (int p = t; p < NPERM; p += 128) sum += expf(row[p] - rowmax);
    sdata[t] = sum; __syncthreads();
    for (int s = 64; s > 0; s >>= 1) {
        if (t < s) sdata[t] += sdata[t + s];
        __syncthreads();
    }

    if (t == 0) {
        float nll = rowmax + logf(sdata[0]) - row[0];
        const float* g = G + (size_t)b * 64;
        float acc = 0.f;
#pragma unroll
        for (int tt = 0; tt < T_SZ - 1; ++tt)
            acc += g[tt * 8 + tt] - 2.f * g[tt * 8 + tt + 1] + g[(tt + 1) * 8 + tt + 1];
        perB[b] = nll + ALPHA * 0.5f * acc;
    }
}

// ---------------------------------------------------------------------------
// Kernel 5: deterministic mean over the 256 per-batch values -> scalar out.
// ---------------------------------------------------------------------------
__global__ void final_kernel(const float* __restrict__ perB, float* __restrict__ out) {
    __shared__ float sdata[B_SZ];
    int t = threadIdx.x;
    sdata[t] = perB[t];
    __syncthreads();
    for (int s = B_SZ / 2; s > 0; s >>= 1) {
        if (t < s) sdata[t] += sdata[t + s];
        __syncthreads();
    }
    if (t == 0) out[0] = sdata[0] * (1.0f / B_SZ);
}

extern "C" void kernel_launch(void* const* d_in, const int* in_sizes, int n_in,
                              void* d_out, int out_size, void* d_ws, size_t ws_size,
                              hipStream_t stream) {
    const float* Z    = (const float*)d_in[0];   // [256,8,128] f32
    const int*   perm = (const int*)d_in[1];     // [1000*8] i32

    // Workspace carve-out (all offsets 256B-aligned enough for float4 use):
    char*  ws     = (char*)d_ws;
    float* G      = (float*)ws;                                   // 256*64   =   64 KB
    float* Q      = (float*)(ws + 65536);                         // 1008*64  =  252 KB
    float* logits = (float*)(ws + 65536 + 258048);                // 256*1008 = 1008 KB
    float* perB   = (float*)(ws + 65536 + 258048 + 1032192);      // 256      =    1 KB

    gram_kernel       <<<B_SZ, 64, 0, stream>>>(Z, G);
    build_q_kernel    <<<NPERM_PAD, 64, 0, stream>>>(perm, Q);
    logits_wmma_kernel<<<dim3(NPERM_PAD / 16, B_SZ / 16), 32, 0, stream>>>(G, Q, logits);
    rowreduce_kernel  <<<B_SZ, 128, 0, stream>>>(logits, G, perB);
    final_kernel      <<<1, B_SZ, 0, stream>>>(perB, (float*)d_out);
}